// RecformerPooler_35725537968809
// MI455X (gfx1250) — compile-verified
//
#include <hip/hip_runtime.h>
#include <hip/hip_bf16.h>

typedef __attribute__((ext_vector_type(2))) float v2f;
typedef __attribute__((ext_vector_type(8))) float v8f;

#define BATCH 64
#define SEQ   1024
#define HID   768
#define NSEG  96          // 3 attrs * 32 items
#define A_MAX 3
#define I_MAX 32
#define CHUNKS_PER_BLOCK 8   // 8 waves, one 16-wide H chunk each
#define BLOCKS_PER_BATCH 6   // 48 chunks / 8
#define TILE_T 32            // tokens staged per tile
#define NTILES (SEQ / TILE_T)

// Segment-mean pooling as a one-hot WMMA matmul:
//   S[96, H] = OneHot[96, L] x X[L, H], then S / counts  (0/0 -> NaN == nanmean).
// One wave per 16-wide H chunk. hidden_states is streamed HBM->LDS with
// gfx1250 async copies (GLOBAL_LOAD_ASYNC_TO_LDS_B128, ASYNCcnt) into a
// double-buffered 2KB tile per wave, overlapping the next tile's fetch with
// the current tile's V_WMMA_F32_16X16X4_F32 work (f32 datapath, exact).
__global__ __launch_bounds__(256)
void recformer_pool_wmma(const float* __restrict__ hidden,
                         const int*  __restrict__ attr,
                         const int*  __restrict__ item,
                         float*      __restrict__ out) {
    __shared__ int   s_seg[SEQ];                          // 4 KB
    __shared__ float s_cnt[NSEG];
    __shared__ float s_stage[8][2][TILE_T * 16];          // 32 KB, 2KB/buf/wave

    const int tid  = threadIdx.x;
    const int b    = blockIdx.x / BLOCKS_PER_BATCH;
    const int cblk = blockIdx.x % BLOCKS_PER_BATCH;

    // ---- Phase 1: per-batch segment ids + counts (shared across 8 waves) ----
    if (tid < NSEG) s_cnt[tid] = 0.0f;
    __syncthreads();

    const int* attr_b = attr + b * SEQ;
    const int* item_b = item + b * SEQ;
    #pragma unroll
    for (int i = tid; i < SEQ; i += 256) {
        int a  = attr_b[i];
        int it = item_b[i];
        bool valid = (a >= 1) & (a <= A_MAX) & (it >= 1) & (it <= I_MAX);
        int s = valid ? (a - 1) * I_MAX + (it - 1) : -1;
        s_seg[i] = s;
        if (valid) atomicAdd(&s_cnt[s], 1.0f);   // ds_add_f32
    }
    __syncthreads();

    // ---- Phase 2: async-staged WMMA one-hot segment sum ----
    const int wave  = tid >> 5;                     // 0..7
    const int lane  = tid & 31;
    const int half  = lane >> 4;                    // 0: lanes 0-15, 1: lanes 16-31
    const int nl    = lane & 15;                    // N (and M) sub-index
    const int kOff  = half * 2;                     // K offset per ISA A/B layout
    const int hbase = (cblk * CHUNKS_PER_BLOCK + wave) * 16;

    const float* Xb = hidden + (size_t)b * SEQ * HID + hbase;

    // async-copy lane mapping: 4 lanes x 16B cover one 64B token row,
    // 8 token rows per b128 instruction, 4 instructions per 32-token tile.
    const int rowIn = lane >> 2;        // 0..7  : token row within instruction
    const int colq  = lane & 3;         // 0..3  : 16B quarter of the 64B row
    // low 32 bits of a generic LDS pointer == DS byte address
    const unsigned stg = (unsigned)(size_t)(&s_stage[wave][0][0]);

    #define ASYNC_TILE(TB, LDSBASE)                                              \
        do {                                                                     \
            _Pragma("unroll")                                                    \
            for (int j = 0; j < 4; ++j) {                                        \
                int tok = (TB) + j * 8 + rowIn;                                  \
                const char* g = (const char*)Xb + (size_t)tok * (HID * 4)        \
                                + colq * 16;                                     \
                unsigned l = (LDSBASE) + (unsigned)((j * 8 + rowIn) * 64         \
                                + colq * 16);                                    \
                asm volatile("global_load_async_to_lds_b128 %0, %1, off"         \
                             :: "v"(l), "v"(g) : "memory");                      \
            }                                                                    \
        } while (0)

    v8f acc0 = {}, acc1 = {}, acc2 = {}, acc3 = {}, acc4 = {}, acc5 = {};

    ASYNC_TILE(0, stg);   // prologue: tile 0 -> buffer 0

    for (int t = 0; t < NTILES; ++t) {
        const int buf = t & 1;
        if (t + 1 < NTILES) {
            ASYNC_TILE((t + 1) * TILE_T, stg + (buf ^ 1) * (TILE_T * 16 * 4));
            asm volatile("s_wait_asynccnt 4" ::: "memory");  // tile t landed
        } else {
            asm volatile("s_wait_asynccnt 0" ::: "memory");
        }

        const float* sb = &s_stage[wave][buf][0];
        const int tb = t * TILE_T;

        #pragma unroll
        for (int s = 0; s < TILE_T; s += 4) {
            const int tA = s + kOff;     // this half-wave's 2 tokens: tA, tA+1
            int2 sp = *(const int2*)&s_seg[tb + tA];      // ds_load_b64
            const int sA = sp.x, sB = sp.y;

            // B tile (4x16, KxN): VGPR0 = rows K=0/2, VGPR1 = rows K=1/3
            v2f Bv;
            Bv.x = sb[tA * 16 + nl];
            Bv.y = sb[(tA + 1) * 16 + nl];

            #pragma unroll
            for (int mi = 0; mi < 6; ++mi) {
                const int mg = mi * 16 + nl;   // this lane's M row (segment id)
                v2f Av;
                Av.x = (sA == mg) ? 1.0f : 0.0f;
                Av.y = (sB == mg) ? 1.0f : 0.0f;
                switch (mi) {
                case 0: acc0 = __builtin_amdgcn_wmma_f32_16x16x4_f32(false, Av, false, Bv, (short)0, acc0, false, false); break;
                case 1: acc1 = __builtin_amdgcn_wmma_f32_16x16x4_f32(false, Av, false, Bv, (short)0, acc1, false, false); break;
                case 2: acc2 = __builtin_amdgcn_wmma_f32_16x16x4_f32(false, Av, false, Bv, (short)0, acc2, false, false); break;
                case 3: acc3 = __builtin_amdgcn_wmma_f32_16x16x4_f32(false, Av, false, Bv, (short)0, acc3, false, false); break;
                case 4: acc4 = __builtin_amdgcn_wmma_f32_16x16x4_f32(false, Av, false, Bv, (short)0, acc4, false, false); break;
                case 5: acc5 = __builtin_amdgcn_wmma_f32_16x16x4_f32(false, Av, false, Bv, (short)0, acc5, false, false); break;
                }
            }
        }
    }
    #undef ASYNC_TILE

    // ---- Phase 3: divide by counts (0/0 -> NaN == nanmean) and store ----
    // C/D layout: VGPR r -> M = r + 8*half; lane nl -> N.
    #pragma unroll
    for (int mi = 0; mi < 6; ++mi) {
        v8f acc = (mi == 0) ? acc0 : (mi == 1) ? acc1 : (mi == 2) ? acc2
                : (mi == 3) ? acc3 : (mi == 4) ? acc4 : acc5;
        #pragma unroll
        for (int r = 0; r < 8; ++r) {
            const int seg = mi * 16 + r + 8 * half;
            const float c = s_cnt[seg];
            out[((size_t)(b * NSEG + seg)) * HID + hbase + nl] = acc[r] / c;
        }
    }
}

extern "C" void kernel_launch(void* const* d_in, const int* in_sizes, int n_in,
                              void* d_out, int out_size, void* d_ws, size_t ws_size,
                              hipStream_t stream) {
    // inputs: [0] attention_mask (int, unused by reference body),
    //         [1] hidden_states (float), [2] attr_type_ids (int),
    //         [3] item_position_ids (int)
    const float* hidden = (const float*)d_in[1];
    const int*   attr   = (const int*)d_in[2];
    const int*   item   = (const int*)d_in[3];
    float*       out    = (float*)d_out;
    (void)in_sizes; (void)n_in; (void)out_size; (void)d_ws; (void)ws_size;

    dim3 grid(BATCH * BLOCKS_PER_BATCH);   // 384 blocks
    dim3 block(256);                       // 8 wave32 waves
    recformer_pool_wmma<<<grid, block, 0, stream>>>(hidden, attr, item, out);
}